// Pooling_41188736369108
// MI455X (gfx1250) — compile-verified
//
#include <hip/hip_runtime.h>
#include <hip/hip_bf16.h>
#include <math.h>

// Problem dims from the reference setup_inputs()
constexpr int B = 8, C = 256, N = 2048, K = 24;

typedef __attribute__((ext_vector_type(2))) float v2f;
typedef __attribute__((ext_vector_type(8))) float v8f;

__device__ __forceinline__ float gelu_exact(float y) {
    return 0.5f * y * (1.0f + erff(y * 0.70710678118654752f));
}

// ---------------------------------------------------------------------------
// Kernel 1: streaming pass over x [B,C,N,K].
//   One block per (b,c) slab (2048 blocks). 8 waves/block, 16 rows/wave/iter.
//   Row sums over K=24 via 6 chained V_WMMA_F32_16X16X4_F32 with B = ones:
//     A (16x4 f32): lane l holds row m=l%16, K = 2*(l/16)+{0,1}  (ISA 7.12.2)
//     D (16x16 f32): vgpr v on lane<16 = rowsum[v], on lane>=16 = rowsum[v+8],
//                    replicated across the 16 columns of each half.
//   lc = 2*mean = rowsum/12 is written to d_out (reused as scratch).
//   Channel partial sums: summing all 32 lanes' 8 accumulators counts every
//   row exactly 16x -> deterministic block tree-reduce, divide by 16, store
//   per-(c,b) partials in d_ws (no atomics).
// ---------------------------------------------------------------------------
__global__ __launch_bounds__(256) void rowmean_wmma_kernel(
    const float* __restrict__ x, float* __restrict__ lc,
    float* __restrict__ ps, float* __restrict__ psq) {
  const int s    = blockIdx.x;         // s = b*C + c
  const int tid  = threadIdx.x;
  const int lane = tid & 31;
  const int wave = tid >> 5;           // 0..7
  const int m    = lane & 15;          // row within 16-row group
  const int h    = lane >> 4;          // half: K-pair select

  const size_t slab  = (size_t)s * N * K;   // float offset of this (b,c) slab
  const size_t slabr = (size_t)s * N;       // row offset

  const v2f bones = {1.0f, 1.0f};           // all-ones B: layout-irrelevant
  float p1 = 0.0f, p2 = 0.0f;               // per-thread partial sum / sumsq

  for (int j = 0; j < N / 128; ++j) {       // 16 iterations: 128 rows/block/iter
    const int rowbase = j * 128 + wave * 16;
    const float* rp = x + slab + (size_t)(rowbase + m) * K + (h << 1);

    // Prefetch next iteration's tile while this one computes.
    if (j + 1 < N / 128) __builtin_prefetch((const void*)(rp + 128 * K), 0, 1);

    v8f acc = {};
#pragma unroll
    for (int i = 0; i < 6; ++i) {           // K chunks of 4: 6 * 4 = 24
      v2f a = *(const v2f*)(rp + 4 * i);    // 8B-aligned (row stride 96B)
      acc = __builtin_amdgcn_wmma_f32_16x16x4_f32(
          /*neg_a=*/false, a, /*neg_b=*/false, bones,
          /*c_mod=*/(short)0, acc, /*reuse_a=*/false, /*reuse_b=*/false);
    }

    const float sc = 1.0f / 12.0f;          // lc = 2*sum/24 = sum/12
    float v[8];
#pragma unroll
    for (int i = 0; i < 8; ++i) {
      v[i] = acc[i] * sc;
      p1 += v[i];
      p2 += v[i] * v[i];
    }
    if (m == 0) {                           // lanes 0 & 16 own unique data
      float4* dst = (float4*)(lc + slabr + (size_t)rowbase + (h << 3));
      dst[0] = make_float4(v[0], v[1], v[2], v[3]);
      dst[1] = make_float4(v[4], v[5], v[6], v[7]);
    }
  }

  // Deterministic block reduction of channel partials (each row counted 16x).
  __shared__ float sb[512];
  sb[tid] = p1;
  sb[256 + tid] = p2;
  __syncthreads();
  for (int off = 128; off > 0; off >>= 1) {
    if (tid < off) {
      sb[tid] += sb[tid + off];
      sb[256 + tid] += sb[256 + tid + off];
    }
    __syncthreads();
  }
  if (tid == 0) {
    const int c = s & (C - 1);
    const int b = s >> 8;                   // s / C
    ps[c * B + b]  = sb[0]   * (1.0f / 16.0f);
    psq[c * B + b] = sb[256] * (1.0f / 16.0f);
  }
}

// ---------------------------------------------------------------------------
// Kernel 2: per-channel BN (batch stats over B*N, biased var) + exact GELU,
// in place on d_out. One block per (b,c); partials summed in fixed order.
// ---------------------------------------------------------------------------
__global__ __launch_bounds__(256) void bn_gelu_kernel(
    float* __restrict__ io, const float* __restrict__ ps,
    const float* __restrict__ psq, const float* __restrict__ gamma,
    const float* __restrict__ beta) {
  const int s = blockIdx.x;
  const int c = s & (C - 1);
  const int tid = threadIdx.x;

  float sum = 0.0f, sq = 0.0f;
#pragma unroll
  for (int b = 0; b < B; ++b) {             // fixed order -> deterministic
    sum += ps[c * B + b];
    sq  += psq[c * B + b];
  }
  const float inv  = 1.0f / (float)(B * N);
  const float mean = sum * inv;
  const float var  = sq * inv - mean * mean;
  const float g    = gamma[c] * rsqrtf(var + 1e-5f);
  const float be   = beta[c];

  float* base = io + (size_t)s * N;
#pragma unroll
  for (int half = 0; half < 2; ++half) {    // 2048 vals: 2 x (256 thr * float4)
    float4* p = (float4*)(base + half * 1024) + tid;
    float4 v = *p;
    v.x = gelu_exact((v.x - mean) * g + be);
    v.y = gelu_exact((v.y - mean) * g + be);
    v.z = gelu_exact((v.z - mean) * g + be);
    v.w = gelu_exact((v.w - mean) * g + be);
    *p = v;
  }
}

extern "C" void kernel_launch(void* const* d_in, const int* in_sizes, int n_in,
                              void* d_out, int out_size, void* d_ws, size_t ws_size,
                              hipStream_t stream) {
  const float* x     = (const float*)d_in[0];   // [B,C,N,K] f32
  const float* gamma = (const float*)d_in[1];   // [C]
  const float* beta  = (const float*)d_in[2];   // [C]
  float* out = (float*)d_out;                   // [B,C,N] f32 (also lc scratch)
  float* ps  = (float*)d_ws;                    // [C*B] partial sums
  float* psq = ps + C * B;                      // [C*B] partial sumsqs

  rowmean_wmma_kernel<<<B * C, 256, 0, stream>>>(x, out, ps, psq);
  bn_gelu_kernel<<<B * C, 256, 0, stream>>>(out, ps, psq, gamma, beta);
}